// OctVolSynth_5815385718784
// MI455X (gfx1250) — compile-verified
//
#include <hip/hip_runtime.h>
#include <stdint.h>

// Volume geometry from the reference.
#define DD 256
#define HH 512
#define WW 512
#define NTOT ((size_t)DD * HH * WW)   // 67,108,864 voxels
#define NLAB 256                      // LUT entries (1 KB)

typedef int   v4i __attribute__((ext_vector_type(4)));
typedef float v4f __attribute__((ext_vector_type(4)));

// Bandwidth-bound streaming kernel:
//  - b128 non-temporal loads/stores (read-once / write-once streams)
//  - LUT staged into LDS with gfx1250 async-copy (global_load_async_to_lds_b128)
//  - per-voxel gather served by the DS pipe (ds_load_b32), VMEM pipe stays
//    dedicated to the three streaming volumes + two output streams.
__global__ __launch_bounds__(256) void octvol_synth_kernel(
    const v4i* __restrict__ labels4,
    const v4f* __restrict__ par4,
    const v4f* __restrict__ tex4,
    const float* __restrict__ lut,      // 256 floats in global
    v4f* __restrict__ out_final4,
    v4f* __restrict__ out_mask4,
    unsigned nvec)                      // NTOT/4
{
    __shared__ float s_lut[NLAB];

    const unsigned tid = threadIdx.x;

    // ---- Stage the 1KB LUT into LDS via CDNA5 async copy ----
    // First 64 lanes (waves 0 and 1, EXEC all-ones within each) each pull
    // 16 bytes: 64 * 16B = 1024B = 256 floats.
    if (tid < 64u) {
        // Flat LDS aperture is 2^32-aligned, so the low 32 bits of the
        // generic address of a __shared__ object ARE the LDS byte offset.
        uint32_t lds_off = (uint32_t)(uintptr_t)(&s_lut[0]) + tid * 16u;
        uint32_t voff    = tid * 16u;  // byte offset into the global LUT
        asm volatile(
            "global_load_async_to_lds_b128 %0, %1, %2\n"
            :
            : "v"(lds_off), "v"(voff), "s"(lut)
            : "memory");
    }
    // ASYNCcnt is per-wave; waves that issued nothing pass through instantly.
    asm volatile("s_wait_asynccnt 0" ::: "memory");
    __syncthreads();

    const unsigned i = blockIdx.x * blockDim.x + tid;   // vec4 index
    if (i >= nvec) return;

    // ---- Streaming reads: 3 x b128 non-temporal ----
    v4i lb = __builtin_nontemporal_load(&labels4[i]);
    v4f pr = __builtin_nontemporal_load(&par4[i]);
    v4f tx = __builtin_nontemporal_load(&tex4[i]);

    // ---- LDS gather + elementwise math (exactly mirrors the reference) ----
    v4f vz;
    vz.x = s_lut[lb.x & (NLAB - 1)];
    vz.y = s_lut[lb.y & (NLAB - 1)];
    vz.z = s_lut[lb.z & (NLAB - 1)];
    vz.w = s_lut[lb.w & (NLAB - 1)];

    v4f vv = vz * tx;                         // vessels * vessel_texture
    vv.x = (vv.x == 0.0f) ? 1.0f : vv.x;      // vessels[vessels==0] = 1
    vv.y = (vv.y == 0.0f) ? 1.0f : vv.y;
    vv.z = (vv.z == 0.0f) ? 1.0f : vv.z;
    vv.w = (vv.w == 0.0f) ? 1.0f : vv.w;

    v4f fin = pr * vv;                        // parenchyma * vessels

    v4f mk;
    mk.x = (lb.x != 0) ? 1.0f : 0.0f;         // vessel_mask = labels.astype(bool)
    mk.y = (lb.y != 0) ? 1.0f : 0.0f;
    mk.z = (lb.z != 0) ? 1.0f : 0.0f;
    mk.w = (lb.w != 0) ? 1.0f : 0.0f;

    // ---- Streaming writes: 2 x b128 non-temporal ----
    __builtin_nontemporal_store(fin, &out_final4[i]);
    __builtin_nontemporal_store(mk,  &out_mask4[i]);
}

extern "C" void kernel_launch(void* const* d_in, const int* in_sizes, int n_in,
                              void* d_out, int out_size, void* d_ws, size_t ws_size,
                              hipStream_t stream) {
    // setup_inputs() order: vessel_labels(i32), parenchyma(f32),
    //                       vessel_texture(f32), label_lut(f32[256])
    const v4i* labels4 = (const v4i*)d_in[0];
    const v4f* par4    = (const v4f*)d_in[1];
    const v4f* tex4    = (const v4f*)d_in[2];
    const float* lut   = (const float*)d_in[3];

    // Outputs concatenated in return order: final_volume then vessel_mask.
    float* out_final = (float*)d_out;
    float* out_mask  = out_final + NTOT;

    const unsigned nvec = (unsigned)(NTOT / 4);           // 16,777,216
    dim3 block(256);
    dim3 grid((nvec + block.x - 1) / block.x);            // 65,536 blocks

    hipLaunchKernelGGL(octvol_synth_kernel, grid, block, 0, stream,
                       labels4, par4, tex4, lut,
                       (v4f*)out_final, (v4f*)out_mask, nvec);
}